// NodeLevelAttentionImproved_78142634983557
// MI455X (gfx1250) — compile-verified
//
#include <hip/hip_runtime.h>
#include <hip/hip_bf16.h>
#include <math.h>

#define N_NODES 20000
#define M_NB 32
#define IN_DIM 256
#define OUT_DIM 256
#define H_HEADS 4
#define D_HEAD 64

typedef __attribute__((ext_vector_type(2))) float v2f;
typedef __attribute__((ext_vector_type(8))) float v8f;

// ---------------------------------------------------------------------------
// Kernel 1: h_proj = h @ W via V_WMMA_F32_16X16X4_F32 (full fp32 precision).
// One wave computes a 16x64 tile: one shared A fragment, 4 B fragments,
// 4 accumulators. K-loop: 64 steps of K=4 -> 256 wmma per wave.
// A layout (16x4 f32): lanes 0-15 -> K={0,1}, lanes 16-31 -> K={2,3}
// B layout (4x16 f32): mirrors A in K; N = lane%16
// C layout: VGPR i -> row i (lanes 0-15), row i+8 (lanes 16-31)
// ---------------------------------------------------------------------------
__global__ __launch_bounds__(256) void gat_gemm_wmma(
    const float* __restrict__ h, const float* __restrict__ W,
    float* __restrict__ h_proj)
{
  const int wave_g = (blockIdx.x * blockDim.x + threadIdx.x) >> 5;
  const int lane   = threadIdx.x & 31;
  const int rt   = wave_g >> 2;      // 1250 row tiles of 16
  const int cg   = wave_g & 3;       // 4 col groups of 64
  const int row0 = rt * 16;
  const int col0 = cg * 64;
  const int half = lane >> 4;        // 0 = lanes 0-15, 1 = lanes 16-31
  const int lm   = lane & 15;

  const float* Arow = h + (size_t)(row0 + lm) * IN_DIM + 2 * half;
  v8f c0 = {}, c1 = {}, c2 = {}, c3 = {};

  for (int k = 0; k < IN_DIM; k += 4) {
    v2f a;
    a.x = Arow[k];
    a.y = Arow[k + 1];
    const float* Brow0 = W + (size_t)(k + 2 * half) * OUT_DIM + col0 + lm;
    const float* Brow1 = Brow0 + OUT_DIM;
    v2f b0; b0.x = Brow0[0];  b0.y = Brow1[0];
    v2f b1; b1.x = Brow0[16]; b1.y = Brow1[16];
    v2f b2; b2.x = Brow0[32]; b2.y = Brow1[32];
    v2f b3; b3.x = Brow0[48]; b3.y = Brow1[48];
    c0 = __builtin_amdgcn_wmma_f32_16x16x4_f32(false, a, false, b0, (short)0, c0, false, false);
    c1 = __builtin_amdgcn_wmma_f32_16x16x4_f32(false, a, false, b1, (short)0, c1, false, false);
    c2 = __builtin_amdgcn_wmma_f32_16x16x4_f32(false, a, false, b2, (short)0, c2, false, false);
    c3 = __builtin_amdgcn_wmma_f32_16x16x4_f32(false, a, false, b3, (short)0, c3, false, false);
  }

  float* Crow = h_proj + (size_t)(row0 + half * 8) * OUT_DIM + col0 + lm;
#pragma unroll
  for (int i = 0; i < 8; ++i) {
    Crow[(size_t)i * OUT_DIM + 0]  = c0[i];
    Crow[(size_t)i * OUT_DIM + 16] = c1[i];
    Crow[(size_t)i * OUT_DIM + 32] = c2[i];
    Crow[(size_t)i * OUT_DIM + 48] = c3[i];
  }
}

// ---------------------------------------------------------------------------
// Kernel 2: el[n,h] = <h_proj[n,h,:], a_l[h,:]>, er likewise.
// One wave per node; lane l holds d = l*8 .. l*8+7 (each lane's 8 elements
// are inside one head since 8 lanes * 8 = 64 = D). Shuffle-reduce groups of 8.
// ---------------------------------------------------------------------------
__global__ __launch_bounds__(256) void gat_scores(
    const float* __restrict__ h_proj, const float* __restrict__ a_l,
    const float* __restrict__ a_r, float* __restrict__ el,
    float* __restrict__ er)
{
  const int node = (blockIdx.x * blockDim.x + threadIdx.x) >> 5;
  const int lane = threadIdx.x & 31;
  const float* hp = h_proj + (size_t)node * OUT_DIM + lane * 8;
  float4 x0 = *(const float4*)hp;
  float4 x1 = *(const float4*)(hp + 4);

  const int head = lane >> 3;
  const int dofs = (lane & 7) * 8;
  const float* al = a_l + head * D_HEAD + dofs;
  const float* ar = a_r + head * D_HEAD + dofs;
  float4 l0 = *(const float4*)al, l1 = *(const float4*)(al + 4);
  float4 r0 = *(const float4*)ar, r1 = *(const float4*)(ar + 4);

  float sl = x0.x*l0.x + x0.y*l0.y + x0.z*l0.z + x0.w*l0.w
           + x1.x*l1.x + x1.y*l1.y + x1.z*l1.z + x1.w*l1.w;
  float sr = x0.x*r0.x + x0.y*r0.y + x0.z*r0.z + x0.w*r0.w
           + x1.x*r1.x + x1.y*r1.y + x1.z*r1.z + x1.w*r1.w;

  for (int off = 1; off < 8; off <<= 1) {
    sl += __shfl_xor(sl, off, 32);
    sr += __shfl_xor(sr, off, 32);
  }
  if ((lane & 7) == 0) {
    el[(size_t)node * H_HEADS + head] = sl;
    er[(size_t)node * H_HEADS + head] = sr;
  }
}

// ---------------------------------------------------------------------------
// Kernel 3: attention softmax + gathered aggregation + residual + GELU + LN.
// One block (256 threads) per node. Aggregation reads per-neighbor rows of
// h_proj contiguously (coalesced; h_proj fits in L2).
// ---------------------------------------------------------------------------
__global__ __launch_bounds__(256) void gat_attn(
    const float* __restrict__ h_proj, const float* __restrict__ el,
    const float* __restrict__ er, const int* __restrict__ nidx,
    const int* __restrict__ nmask, const float* __restrict__ gamma,
    const float* __restrict__ beta, float* __restrict__ out)
{
  __shared__ int   s_idx[M_NB];
  __shared__ float s_alpha[H_HEADS][M_NB];
  __shared__ float s_el[H_HEADS];
  __shared__ float s_rs[H_HEADS];
  __shared__ float s_red1[8], s_red2[8];
  __shared__ float s_mu, s_rstd;

  const int n = blockIdx.x;
  const int tid = threadIdx.x;

  if (tid < M_NB)    s_idx[tid] = nidx[(size_t)n * M_NB + tid];
  if (tid < H_HEADS) s_el[tid]  = el[(size_t)n * H_HEADS + tid];
  __syncthreads();

  if (tid < M_NB * H_HEADS) {
    const int m = tid >> 2, hh = tid & 3;
    float v = s_el[hh] + er[(size_t)s_idx[m] * H_HEADS + hh];
    v = (v > 0.f) ? v : 0.2f * v;                 // leaky_relu(0.2)
    if (nmask[(size_t)n * M_NB + m] == 0) v = -1.0e9f;
    s_alpha[hh][m] = v;
  }
  __syncthreads();

  if (tid < H_HEADS) {
    float mx = -3.4e38f;
    for (int m = 0; m < M_NB; ++m) mx = fmaxf(mx, s_alpha[tid][m]);
    float sum = 0.f;
    for (int m = 0; m < M_NB; ++m) {
      float ex = __expf(s_alpha[tid][m] - mx);
      s_alpha[tid][m] = ex;
      sum += ex;
    }
    s_rs[tid] = 1.0f / sum;
  }
  __syncthreads();

  const int hh = tid >> 6;                        // head = d / 64
  const float rs = s_rs[hh];
  float acc = 0.f;
#pragma unroll 4
  for (int m = 0; m < M_NB; ++m)
    acc += s_alpha[hh][m] * h_proj[(size_t)s_idx[m] * OUT_DIM + tid];

  float x = acc * rs + h_proj[(size_t)n * OUT_DIM + tid];   // + residual
  float g = 0.5f * x * (1.0f + erff(x * 0.70710678118654752f)); // exact GELU

  // block LayerNorm over 256
  float s1 = g, s2 = g * g;
  for (int off = 16; off > 0; off >>= 1) {
    s1 += __shfl_xor(s1, off, 32);
    s2 += __shfl_xor(s2, off, 32);
  }
  const int wid = tid >> 5, lane = tid & 31;
  if (lane == 0) { s_red1[wid] = s1; s_red2[wid] = s2; }
  __syncthreads();
  if (tid == 0) {
    float t1 = 0.f, t2 = 0.f;
    for (int i = 0; i < 8; ++i) { t1 += s_red1[i]; t2 += s_red2[i]; }
    const float mu  = t1 * (1.0f / OUT_DIM);
    const float var = t2 * (1.0f / OUT_DIM) - mu * mu;
    s_mu = mu;
    s_rstd = rsqrtf(var + 1e-5f);
  }
  __syncthreads();

  out[(size_t)n * OUT_DIM + tid] = (g - s_mu) * s_rstd * gamma[tid] + beta[tid];
}

// ---------------------------------------------------------------------------
extern "C" void kernel_launch(void* const* d_in, const int* in_sizes, int n_in,
                              void* d_out, int out_size, void* d_ws, size_t ws_size,
                              hipStream_t stream) {
  (void)in_sizes; (void)n_in; (void)out_size; (void)ws_size;
  const float* h     = (const float*)d_in[0];
  const int*   nidx  = (const int*)d_in[1];
  const int*   nmask = (const int*)d_in[2];
  const float* W     = (const float*)d_in[3];
  const float* a_l   = (const float*)d_in[4];
  const float* a_r   = (const float*)d_in[5];
  const float* gamma = (const float*)d_in[6];
  const float* beta  = (const float*)d_in[7];
  float* out = (float*)d_out;

  char* ws = (char*)d_ws;
  float* h_proj = (float*)ws;                                     // 20000*256 f32
  float* el = (float*)(ws + (size_t)N_NODES * OUT_DIM * sizeof(float));
  float* er = el + (size_t)N_NODES * H_HEADS;

  dim3 blk(256);
  // 1250 row-tiles * 4 col-groups = 5000 waves / 8 waves-per-block = 625 blocks
  gat_gemm_wmma<<<625, blk, 0, stream>>>(h, W, h_proj);
  // one wave per node
  gat_scores<<<N_NODES / 8, blk, 0, stream>>>(h_proj, a_l, a_r, el, er);
  // one block per node
  gat_attn<<<N_NODES, blk, 0, stream>>>(h_proj, el, er, nidx, nmask, gamma, beta, out);
}